// _ScaledDotProductAttention_11587821764888
// MI455X (gfx1250) — compile-verified
//
#include <hip/hip_runtime.h>
#include <hip/hip_bf16.h>
#include <math.h>

#define B_ 2
#define H_ 16
#define S_ 2048
#define D_ 64

typedef float v2f __attribute__((ext_vector_type(2)));
typedef float v8f __attribute__((ext_vector_type(8)));

#define NEG_INF (-__builtin_inff())

// exp with NaN-safe clamp: fmax drops the NaN from (-inf)-(-inf), and
// expf(-128) underflows to exactly 0.0f, so masked contributions are exact 0.
__device__ __forceinline__ float exp_clamped(float x) {
    return __expf(fmaxf(x, -128.0f));
}

// ---------------------------------------------------------------------------
// Kernel A: scores = QK^T * eff_scale + prev, masked; online softmax stats.
// One wave owns a 16-row strip of one (b,h). WMMA f32 16x16x4.
// Stats kept per-lane (lane owns columns colq+16j), merged across the 16
// lanes of each half once after the loop. Fully branchless hot loop.
// ---------------------------------------------------------------------------
__global__ __launch_bounds__(256) void attn_scores_kernel(
    const float* __restrict__ q, const float* __restrict__ k,
    const float* __restrict__ prev, const unsigned char* __restrict__ kpad,
    const float* __restrict__ scale_p, const float* __restrict__ extra_p,
    float* __restrict__ out_s, float* __restrict__ ws_m, float* __restrict__ ws_l)
{
    const int lane = threadIdx.x & 31;
    const int wave = blockIdx.x * (blockDim.x >> 5) + (threadIdx.x >> 5);
    const int it   = wave & 127;     // row tile within (b,h)
    const int bh   = wave >> 7;      // 0..31
    const int b    = bh >> 4;
    const int row0 = it << 4;
    const int colq = lane & 15;
    const int half = lane >> 4;      // 0: lanes 0-15, 1: lanes 16-31

    const float eff = scale_p[0] * fminf(fmaxf(extra_p[0], 0.01f), 50.0f);

    const size_t bh_sd = (size_t)bh * (S_ * D_);
    const size_t bh_ss = (size_t)bh * S_ * S_;

    // Q strip as 16 WMMA-A fragments (A 16x4 f32: lane<16 -> K{0,1},
    // lane>=16 -> K{2,3}; row = lane&15).
    const float* qrow = q + bh_sd + (size_t)(row0 + colq) * D_ + half * 2;
    v2f aq[16];
#pragma unroll
    for (int kc = 0; kc < 16; ++kc)
        aq[kc] = *reinterpret_cast<const v2f*>(qrow + kc * 4);

    const float* kbh = k + bh_sd;            // [D,S] slice for this (b,h)
    const unsigned char* kp = kpad + b * S_;

    float m_lane[8], l_lane[8];
#pragma unroll
    for (int r = 0; r < 8; ++r) { m_lane[r] = NEG_INF; l_lane[r] = 0.0f; }

    // diag=false: j < it, causality can never mask (col < row0 <= row).
    auto tile = [&](int j, bool diag) {
        const int col = j * 16 + colq;
        v8f acc = {};
        const float* kcol = kbh + (size_t)(half * 2) * S_ + col;
#pragma unroll
        for (int kc = 0; kc < 16; ++kc) {
            v2f bk;                          // B 4x16 layout (rows striped by half)
            bk.x = kcol[(size_t)(kc * 4) * S_];
            bk.y = kcol[(size_t)(kc * 4 + 1) * S_];
            acc = __builtin_amdgcn_wmma_f32_16x16x4_f32(
                false, aq[kc], false, bk, (short)0, acc, false, false);
        }

        const bool pad = (kp[col] != 0);
        const float* prow = prev  + bh_ss + (size_t)(row0 + half * 8) * S_ + col;
        float*       srow = out_s + bh_ss + (size_t)(row0 + half * 8) * S_ + col;

        float pv[8];
#pragma unroll
        for (int r = 0; r < 8; ++r)
            pv[r] = __builtin_nontemporal_load(prow + (size_t)r * S_);

#pragma unroll
        for (int r = 0; r < 8; ++r) {
            const bool dead = diag ? (pad || col > (row0 + half * 8 + r)) : pad;
            const float sv = dead ? NEG_INF : fmaf(acc[r], eff, pv[r]);
            srow[(size_t)r * S_] = sv;       // re-read by kernel B: regular store
            // branchless online softmax update (exact: see exp_clamped)
            const float nm = fmaxf(m_lane[r], sv);
            l_lane[r] = fmaf(l_lane[r], exp_clamped(m_lane[r] - nm),
                             exp_clamped(sv - nm));
            m_lane[r] = nm;
        }
    };

    for (int j = 0; j < it; ++j) tile(j, false);
    tile(it, true);

    // merge (m,l) across the 16 lanes of each half (branchless)
#pragma unroll
    for (int r = 0; r < 8; ++r) {
        float m = m_lane[r], l = l_lane[r];
#pragma unroll
        for (int off = 8; off >= 1; off >>= 1) {
            const float om = __shfl_xor(m, off, 32);
            const float ol = __shfl_xor(l, off, 32);
            const float nm = fmaxf(m, om);
            l = fmaf(l, exp_clamped(m - nm), ol * exp_clamped(om - nm));
            m = nm;
        }
        m_lane[r] = m; l_lane[r] = l;
    }

    // tiles fully masked by causality: pure -inf, streamed (never re-read)
    for (int j = it + 1; j < (S_ / 16); ++j) {
        float* srow = out_s + bh_ss + (size_t)(row0 + half * 8) * S_ + j * 16 + colq;
#pragma unroll
        for (int r = 0; r < 8; ++r)
            __builtin_nontemporal_store(NEG_INF, srow + (size_t)r * S_);
    }

    if (colq == 0) {                         // all lanes hold merged stats
#pragma unroll
        for (int r = 0; r < 8; ++r) {
            ws_m[wave * 16 + half * 8 + r] = m_lane[r];
            ws_l[wave * 16 + half * 8 + r] = l_lane[r];
        }
    }
}

// ---------------------------------------------------------------------------
// Kernel B: weights = exp(scores - m)/l, output = weights @ V.
// Scores re-read (read-once -> NT) in C/D layout for coalesced weight stores;
// tile bounced through LDS to produce WMMA-A fragments for the PV matmul.
// ---------------------------------------------------------------------------
__global__ __launch_bounds__(256) void attn_out_kernel(
    const float* __restrict__ v, const float* __restrict__ scores,
    const float* __restrict__ ws_m, const float* __restrict__ ws_l,
    float* __restrict__ out_w, float* __restrict__ out_o)
{
    __shared__ float lds[8][16 * 18];        // per-wave 16x16 tile, stride 18
    const int lane  = threadIdx.x & 31;
    const int wslot = threadIdx.x >> 5;
    const int wave  = blockIdx.x * (blockDim.x >> 5) + wslot;
    const int it    = wave & 127;
    const int bh    = wave >> 7;
    const int row0  = it << 4;
    const int colq  = lane & 15;
    const int half  = lane >> 4;

    const size_t bh_ss = (size_t)bh * S_ * S_;
    const size_t bh_sd = (size_t)bh * (S_ * D_);

    float mcd[8], linv[8];
#pragma unroll
    for (int r = 0; r < 8; ++r) {
        mcd[r]  = ws_m[wave * 16 + half * 8 + r];
        linv[r] = 1.0f / ws_l[wave * 16 + half * 8 + r];
    }

    v8f acc[4] = {};                         // O tile: 16 rows x 64 cols
    const float* vbh = v + bh_sd;
    float* lw = &lds[wslot][0];

    for (int j = 0; j <= it; ++j) {
        const float* srow = scores + bh_ss + (size_t)(row0 + half * 8) * S_ + j * 16 + colq;
        float*       wrow = out_w  + bh_ss + (size_t)(row0 + half * 8) * S_ + j * 16 + colq;
        float sv[8];
#pragma unroll
        for (int r = 0; r < 8; ++r)
            sv[r] = __builtin_nontemporal_load(srow + (size_t)r * S_);
#pragma unroll
        for (int r = 0; r < 8; ++r) {
            const float w = __expf(sv[r] - mcd[r]) * linv[r]; // exp(-inf)=0 masked
            __builtin_nontemporal_store(w, wrow + (size_t)r * S_);
            lw[(half * 8 + r) * 18 + colq] = w;
        }
        asm volatile("s_wait_dscnt 0" ::: "memory");        // LDS RAW fence (same wave)

        v2f wa[4];                            // W tile as A 16x4 fragments
#pragma unroll
        for (int kc = 0; kc < 4; ++kc) {
            const float* p = lw + colq * 18 + kc * 4 + half * 2;
            wa[kc].x = p[0];
            wa[kc].y = p[1];
        }
#pragma unroll
        for (int kc = 0; kc < 4; ++kc) {
            const float* vr = vbh + (size_t)(j * 16 + kc * 4 + half * 2) * D_ + colq;
#pragma unroll
            for (int nt = 0; nt < 4; ++nt) {
                v2f bv;                       // V rows as B 4x16 fragments
                bv.x = vr[nt * 16];
                bv.y = vr[nt * 16 + D_];
                acc[nt] = __builtin_amdgcn_wmma_f32_16x16x4_f32(
                    false, wa[kc], false, bv, (short)0, acc[nt], false, false);
            }
        }
        asm volatile("" ::: "memory");        // keep LDS reads before next iter's writes
    }

    // causally dead weight tiles are exactly zero
    for (int j = it + 1; j < (S_ / 16); ++j) {
        float* wrow = out_w + bh_ss + (size_t)(row0 + half * 8) * S_ + j * 16 + colq;
#pragma unroll
        for (int r = 0; r < 8; ++r)
            __builtin_nontemporal_store(0.0f, wrow + (size_t)r * S_);
    }

    float* orow = out_o + bh_sd + (size_t)(row0 + half * 8) * D_ + colq;
#pragma unroll
    for (int nt = 0; nt < 4; ++nt)
#pragma unroll
        for (int r = 0; r < 8; ++r)
            orow[(size_t)r * D_ + nt * 16] = acc[nt][r];
}

// ---------------------------------------------------------------------------
extern "C" void kernel_launch(void* const* d_in, const int* in_sizes, int n_in,
                              void* d_out, int out_size, void* d_ws, size_t ws_size,
                              hipStream_t stream) {
    const float*         q     = (const float*)d_in[0];
    const float*         k     = (const float*)d_in[1];
    const float*         v     = (const float*)d_in[2];
    const float*         prev  = (const float*)d_in[3];
    const unsigned char* kpad  = (const unsigned char*)d_in[4];
    // d_in[5] = attn_mask (causal triu) handled analytically
    const float*         scale = (const float*)d_in[6];
    const float*         extra = (const float*)d_in[7];

    float* out   = (float*)d_out;
    float* out_o = out;                                     // [B,H,S,D]
    float* out_w = out_o + (size_t)B_ * H_ * S_ * D_;       // [B,H,S,S]
    float* out_s = out_w + (size_t)B_ * H_ * S_ * S_;       // [B,H,S,S]

    float* ws_m = (float*)d_ws;                             // B*H*S row maxes
    float* ws_l = ws_m + (size_t)B_ * H_ * S_;              // B*H*S row sums

    const int waves  = B_ * H_ * (S_ / 16);                 // 4096
    const dim3 block(256);                                  // 8 waves/block
    const dim3 grid(waves / 8);                             // 512

    attn_scores_kernel<<<grid, block, 0, stream>>>(q, k, prev, kpad, scale, extra,
                                                   out_s, ws_m, ws_l);
    attn_out_kernel<<<grid, block, 0, stream>>>(v, out_s, ws_m, ws_l, out_w, out_o);
}